// LocationEncoder_9320079033049
// MI455X (gfx1250) — compile-verified
//
#include <hip/hip_runtime.h>
#include <hip/hip_bf16.h>

#define N_PED 16384
#define NTILES (N_PED / 16)          // 1024 column tiles of width 16
#define WAVES_PER_BLOCK 8
#define TILES_PER_WAVE (NTILES / WAVES_PER_BLOCK)  // 128

typedef __attribute__((ext_vector_type(2))) float v2f;
typedef __attribute__((ext_vector_type(8))) float v8f;

__device__ __forceinline__ float fast_exp(float x) {
  // exp(x) = exp2(x * log2(e)); single v_exp_f32
  return __builtin_amdgcn_exp2f(x * 1.4426950408889634f);
}

// ---------------------------------------------------------------------------
// Phase 1: tiny per-row MLP, store h transposed+pair-interleaved:
//   hp[p * N + n] = { h[n][2p], h[n][2p+1] }   (p = 0..31)
// This makes every WMMA f32 A/B fragment load a coalesced b64 load.
// ---------------------------------------------------------------------------
__global__ void __launch_bounds__(256) le_mlp_kernel(
    const float* __restrict__ data,
    const float* __restrict__ W1, const float* __restrict__ b1,
    const float* __restrict__ W2, const float* __restrict__ b2,
    const float* __restrict__ W3, const float* __restrict__ b3,
    v2f* __restrict__ hp) {
  const int n = blockIdx.x * blockDim.x + threadIdx.x;
  const float x0 = data[2 * n];
  const float x1 = data[2 * n + 1];

  float h1[32];
#pragma unroll
  for (int j = 0; j < 32; ++j) {
    float a = fmaf(x0, W1[j], fmaf(x1, W1[32 + j], b1[j]));
    h1[j] = fmaxf(a, 0.0f);
  }

  float h2[64];
#pragma unroll
  for (int j = 0; j < 64; ++j) h2[j] = b2[j];
#pragma unroll
  for (int k = 0; k < 32; ++k) {
#pragma unroll
    for (int j = 0; j < 64; ++j) h2[j] = fmaf(h1[k], W2[k * 64 + j], h2[j]);
  }
#pragma unroll
  for (int j = 0; j < 64; ++j) h2[j] = fmaxf(h2[j], 0.0f);

#pragma unroll
  for (int p = 0; p < 32; ++p) {
    float o0 = b3[2 * p];
    float o1 = b3[2 * p + 1];
#pragma unroll
    for (int k = 0; k < 64; ++k) {
      o0 = fmaf(h2[k], W3[k * 64 + 2 * p], o0);
      o1 = fmaf(h2[k], W3[k * 64 + 2 * p + 1], o1);
    }
    v2f o = {o0, o1};
    hp[(size_t)p * N_PED + n] = o;
  }
}

// ---------------------------------------------------------------------------
// Phase 2: per 16-row block, fused  softmax(h @ h^T)  via V_WMMA_F32_16X16X4.
// Pass 1: lane-local online (max,sum) over 128 column tiles per wave,
//         then butterfly-reduce over 16-lane halves + LDS merge over 8 waves.
// Pass 2: recompute tiles, write exp(v-m)/s with non-temporal stores.
// ---------------------------------------------------------------------------
__global__ void __launch_bounds__(256) le_gram_softmax_kernel(
    const v2f* __restrict__ hp, float* __restrict__ out) {
  const int lane = threadIdx.x & 31;
  const int wave = threadIdx.x >> 5;
  const int half = lane >> 4;     // 0: K%4 in {0,1} / rows M=0..7 ; 1: {2,3} / M=8..15
  const int ln   = lane & 15;
  const int rowbase = blockIdx.x * 16;

  __shared__ float lm[WAVES_PER_BLOCK][16];
  __shared__ float ls[WAVES_PER_BLOCK][16];
  __shared__ float fm[16];
  __shared__ float fsinv[16];

  // --- A fragments: a[s] holds h[rowbase+ln][4s + 2*half .. +1]  (16 K-steps)
  v2f a[16];
  const v2f* ap = hp + (size_t)half * N_PED + rowbase + ln;
#pragma unroll
  for (int s = 0; s < 16; ++s) a[s] = ap[(size_t)(2 * s) * N_PED];

  const int ct0 = wave * TILES_PER_WAVE;
  const v2f* bp = hp + (size_t)half * N_PED + ln;

  float m[8], ssum[8];
#pragma unroll
  for (int i = 0; i < 8; ++i) { m[i] = -__builtin_inff(); ssum[i] = 0.0f; }

  // ---------------- pass 1: online row max / sum-of-exp (lane-local) --------
  for (int ct = ct0; ct < ct0 + TILES_PER_WAVE; ++ct) {
    const v2f* bt = bp + ct * 16;
    v8f c0 = {};
    v8f c1 = {};
#pragma unroll
    for (int s = 0; s < 16; s += 2) {
      v2f b0 = bt[(size_t)(2 * s) * N_PED];
      v2f b1 = bt[(size_t)(2 * s + 2) * N_PED];
      c0 = __builtin_amdgcn_wmma_f32_16x16x4_f32(false, a[s],     false, b0,
                                                 (short)0, c0, false, false);
      c1 = __builtin_amdgcn_wmma_f32_16x16x4_f32(false, a[s + 1], false, b1,
                                                 (short)0, c1, false, false);
    }
#pragma unroll
    for (int i = 0; i < 8; ++i) {
      float v  = c0[i] + c1[i];
      float mo = m[i];
      float mn = fmaxf(mo, v);
      ssum[i]  = ssum[i] * fast_exp(mo - mn) + fast_exp(v - mn);
      m[i]     = mn;
    }
  }

  // --- reduce (m,s) across the 16 lanes of each half-group -----------------
#pragma unroll
  for (int i = 0; i < 8; ++i) {
    float mi = m[i], si = ssum[i];
#pragma unroll
    for (int off = 8; off >= 1; off >>= 1) {
      float mo = __shfl_xor(mi, off, 32);
      float so = __shfl_xor(si, off, 32);
      float mn = fmaxf(mi, mo);
      si = si * fast_exp(mi - mn) + so * fast_exp(mo - mn);
      mi = mn;
    }
    m[i] = mi; ssum[i] = si;
  }
  if (ln == 0) {
#pragma unroll
    for (int i = 0; i < 8; ++i) {
      lm[wave][i + half * 8] = m[i];
      ls[wave][i + half * 8] = ssum[i];
    }
  }
  __syncthreads();

  // --- merge across the 8 waves (each covered a different column range) ----
  if (threadIdx.x < 16) {
    const int r = threadIdx.x;
    float mf = lm[0][r];
#pragma unroll
    for (int w = 1; w < WAVES_PER_BLOCK; ++w) mf = fmaxf(mf, lm[w][r]);
    float sf = 0.0f;
#pragma unroll
    for (int w = 0; w < WAVES_PER_BLOCK; ++w)
      sf += ls[w][r] * fast_exp(lm[w][r] - mf);
    fm[r]    = mf;
    fsinv[r] = 1.0f / sf;
  }
  __syncthreads();

  float rm[8], rs[8];
#pragma unroll
  for (int i = 0; i < 8; ++i) {
    rm[i] = fm[i + half * 8];
    rs[i] = fsinv[i + half * 8];
  }

  // ---------------- pass 2: recompute tiles, write softmax (NT stores) -----
  for (int ct = ct0; ct < ct0 + TILES_PER_WAVE; ++ct) {
    const v2f* bt = bp + ct * 16;
    v8f c0 = {};
    v8f c1 = {};
#pragma unroll
    for (int s = 0; s < 16; s += 2) {
      v2f b0 = bt[(size_t)(2 * s) * N_PED];
      v2f b1 = bt[(size_t)(2 * s + 2) * N_PED];
      c0 = __builtin_amdgcn_wmma_f32_16x16x4_f32(false, a[s],     false, b0,
                                                 (short)0, c0, false, false);
      c1 = __builtin_amdgcn_wmma_f32_16x16x4_f32(false, a[s + 1], false, b1,
                                                 (short)0, c1, false, false);
    }
    const int colbase = ct * 16;
#pragma unroll
    for (int i = 0; i < 8; ++i) {
      float v = c0[i] + c1[i];
      float o = fast_exp(v - rm[i]) * rs[i];
      const size_t row = (size_t)(rowbase + i + half * 8);
      __builtin_nontemporal_store(o, &out[row * N_PED + colbase + ln]);
    }
  }
}

// ---------------------------------------------------------------------------
extern "C" void kernel_launch(void* const* d_in, const int* in_sizes, int n_in,
                              void* d_out, int out_size, void* d_ws, size_t ws_size,
                              hipStream_t stream) {
  const float* data = (const float*)d_in[0];
  const float* W1   = (const float*)d_in[1];
  const float* b1   = (const float*)d_in[2];
  const float* W2   = (const float*)d_in[3];
  const float* b2   = (const float*)d_in[4];
  const float* W3   = (const float*)d_in[5];
  const float* b3   = (const float*)d_in[6];
  float* out = (float*)d_out;
  v2f*   hp  = (v2f*)d_ws;  // needs 32 * 16384 * 8 B = 4 MiB of scratch

  le_mlp_kernel<<<N_PED / 256, 256, 0, stream>>>(data, W1, b1, W2, b2, W3, b3, hp);
  le_gram_softmax_kernel<<<NTILES, 256, 0, stream>>>(hp, out);
}